// BasisJastrow_12429635355204
// MI455X (gfx1250) — compile-verified
//
#include <hip/hip_runtime.h>

typedef __attribute__((ext_vector_type(2))) float v2f;
typedef __attribute__((ext_vector_type(8))) float v8f;

#define WAVES_PER_BLOCK 4
#define XS 36   // padded LDS row stride (floats) for X and C: 16B-aligned float4, conflict-free
#define YS 68   // padded LDS row stride (floats) for Y

__global__ __launch_bounds__(128) void jastrow_pairsum_kernel(
    const float* __restrict__ basis,   // [B, 64, 32] f32
    const float* __restrict__ coeff,   // [32*32] f32, C[u,v] = coeff[u*32+v]
    float* __restrict__ out,           // [B] f32
    int B)
{
    __shared__ float sC[32 * XS];                    // C, shared by whole block
    __shared__ float sX[WAVES_PER_BLOCK][64 * XS];   // per-wave X
    __shared__ float sY[WAVES_PER_BLOCK][32 * YS];   // per-wave Y = C * X^T

    const int tid  = threadIdx.x;
    const int wave = tid >> 5;
    const int lane = tid & 31;
    const int half = lane >> 4;   // 0: lanes 0-15, 1: lanes 16-31
    const int l16  = lane & 15;
    const int b    = blockIdx.x * WAVES_PER_BLOCK + wave;

    // ---- cooperative load of C (1024 floats = 256 float4) into LDS ----
    {
        const float4* cg = reinterpret_cast<const float4*>(coeff);
        for (int q = tid; q < 256; q += 128) {
            float4 v = cg[q];
            int e = q * 4;
            int r = e >> 5, c = e & 31;
            *reinterpret_cast<float4*>(&sC[r * XS + c]) = v;
        }
    }

    // ---- per-wave load of X (2048 floats = 512 float4), coalesced ----
    if (b < B) {
        const float4* xg = reinterpret_cast<const float4*>(basis + (size_t)b * 2048);
        float* sx = sX[wave];
        #pragma unroll 4
        for (int q = 0; q < 16; ++q) {
            int f = lane + 32 * q;          // float4 index 0..511
            float4 v = xg[f];
            int e = f * 4;
            int r = e >> 5, c = e & 31;
            *reinterpret_cast<float4*>(&sx[r * XS + c]) = v;
        }
    }
    __syncthreads();   // publishes sC across waves (barrier is block-uniform)

    if (b < B) {
        float* sx = sX[wave];
        float* sy = sY[wave];

        // ---- Stage 1: Y = C * X^T  (32x64), tiles [ut 0..1] x [it 0..3] ----
        for (int ut = 0; ut < 2; ++ut) {
            for (int it = 0; it < 4; ++it) {
                v8f acc = {};
                #pragma unroll
                for (int k0 = 0; k0 < 32; k0 += 4) {
                    // A frag: A[m,k] = C[ut*16+m, k];  lane holds (m=l16, k=k0+2*half+{0,1})
                    v2f a;
                    {
                        const float* p = &sC[(ut * 16 + l16) * XS + k0 + 2 * half];
                        a.x = p[0]; a.y = p[1];
                    }
                    // B frag: B[k,n] = X^T[k, it*16+n] = X[it*16+n, k]
                    v2f bb;
                    {
                        const float* p = &sx[(it * 16 + l16) * XS + k0 + 2 * half];
                        bb.x = p[0]; bb.y = p[1];
                    }
                    acc = __builtin_amdgcn_wmma_f32_16x16x4_f32(
                        false, a, false, bb, (short)0, acc, false, false);
                }
                // D layout: VGPR r, lane -> Y[ut*16 + r + 8*half, it*16 + l16]
                #pragma unroll
                for (int r = 0; r < 8; ++r)
                    sy[(ut * 16 + r + 8 * half) * YS + it * 16 + l16] = acc[r];
            }
        }
        // same-wave LDS RAW: DS ops are in-order; compiler inserts any needed dscnt waits

        // ---- Stage 2: M = X * Y, strict-lower-triangular sum ----
        float part = 0.0f;
        for (int jt = 0; jt < 4; ++jt) {
            for (int it = 0; it <= jt; ++it) {
                v8f acc = {};
                #pragma unroll
                for (int k0 = 0; k0 < 32; k0 += 4) {
                    // A frag: X[jt*16 + l16, k0 + 2*half + {0,1}]
                    v2f a;
                    {
                        const float* p = &sx[(jt * 16 + l16) * XS + k0 + 2 * half];
                        a.x = p[0]; a.y = p[1];
                    }
                    // B frag: Y[k0 + 2*half + c, it*16 + l16]
                    v2f bb;
                    bb.x = sy[(k0 + 2 * half + 0) * YS + it * 16 + l16];
                    bb.y = sy[(k0 + 2 * half + 1) * YS + it * 16 + l16];
                    acc = __builtin_amdgcn_wmma_f32_16x16x4_f32(
                        false, a, false, bb, (short)0, acc, false, false);
                }
                if (jt > it) {
                    #pragma unroll
                    for (int r = 0; r < 8; ++r) part += acc[r];
                } else {
                    // diagonal tile: include only local_row > local_col
                    const int rowoff = 8 * half;
                    #pragma unroll
                    for (int r = 0; r < 8; ++r)
                        if (r + rowoff > l16) part += acc[r];
                }
            }
        }

        // ---- wave32 reduction, lane 0 writes J_b ----
        #pragma unroll
        for (int off = 16; off > 0; off >>= 1)
            part += __shfl_xor(part, off, 32);
        if (lane == 0)
            out[b] = part * (1.0f / 2016.0f);   // P = 64*63/2
    }
}

extern "C" void kernel_launch(void* const* d_in, const int* in_sizes, int n_in,
                              void* d_out, int out_size, void* d_ws, size_t ws_size,
                              hipStream_t stream) {
    const float* basis = (const float*)d_in[0];   // [B,64,32]
    const float* coeff = (const float*)d_in[1];   // [1024]
    float* out = (float*)d_out;                   // [B]
    const int B = in_sizes[0] / (64 * 32);        // 384
    const int grid = (B + WAVES_PER_BLOCK - 1) / WAVES_PER_BLOCK;
    jastrow_pairsum_kernel<<<grid, 128, 0, stream>>>(basis, coeff, out, B);
}